// ApplyHomography_56195352101247
// MI455X (gfx1250) — compile-verified
//
#include <hip/hip_runtime.h>

// ApplyHomography: fused projective-warp + bilinear sample.
// B=4, D=32, C=4, H=256, W=256. Layout (b,d,c,h,w), fp32 in/out.

#define NBD   128          // B*D
#define NC    4
#define NH    256
#define NW    256
#define NHW   (NH * NW)    // 65536, channel stride
#define ROWS  4            // rows per block

__global__ __launch_bounds__(256)
void ApplyHomography_warp_kernel(const float* __restrict__ hmat,
                                 const float* __restrict__ src,
                                 float* __restrict__ out) {
    const int x  = threadIdx.x;              // 0..255 : output column
    const int yb = blockIdx.x * ROWS;        // base output row
    const int bd = blockIdx.y;               // 0..127 : (b,d) slice

    // Per-block uniform homography -> scalar (SMEM) loads.
    const float* __restrict__ hm = hmat + bd * 9;
    const float h0 = hm[0], h1 = hm[1], h2 = hm[2];
    const float h3 = hm[3], h4 = hm[4], h5 = hm[5];
    const float h6 = hm[6], h7 = hm[7], h8 = hm[8];

    const float* __restrict__ sbase = src + (size_t)bd * (NC * NHW);
    float*       __restrict__ obase = out + (size_t)bd * (NC * NHW);

    // Prefetch the rows we are most likely to sample (homography is a small
    // perturbation of identity in this problem; a wrong prefetch is harmless).
#pragma unroll
    for (int c = 0; c < NC; ++c)
        __builtin_prefetch(sbase + c * NHW + yb * NW + x, 0, 3);

    const float xf = (float)x;

#pragma unroll
    for (int r = 0; r < ROWS; ++r) {
        const int   y  = yb + r;
        const float yf = (float)y;

        // Row-constant parts are wave-uniform (SALU/SGPR); 3 FMAs per lane.
        const float cx = fmaf(h1, yf, h2);
        const float cy = fmaf(h4, yf, h5);
        const float cz = fmaf(h7, yf, h8);

        const float X = fmaf(h0, xf, cx);
        const float Y = fmaf(h3, xf, cy);
        const float Z = fmaf(h6, xf, cz);
        const float rz = 1.0f / Z;
        const float gxp = X * rz;             // pixel-space warped coords
        const float gyp = Y * rz;

        // Replicate the reference's normalize -> denormalize chain.
        const float gx = (gxp - 127.5f) * (1.0f / 127.5f);
        const float gy = (gyp - 127.5f) * (1.0f / 127.5f);
        const float ix = ((gx + 1.0f) * (float)NW - 1.0f) * 0.5f;
        const float iy = ((gy + 1.0f) * (float)NH - 1.0f) * 0.5f;

        const float x0f = floorf(ix), y0f = floorf(iy);
        const float wx = ix - x0f,    wy = iy - y0f;

        const int ix0 = (int)x0f, iy0 = (int)y0f;
        const int ix1 = ix0 + 1,  iy1 = iy0 + 1;

        const bool okx0 = (ix0 >= 0) & (ix0 < NW);
        const bool okx1 = (ix1 >= 0) & (ix1 < NW);
        const bool oky0 = (iy0 >= 0) & (iy0 < NH);
        const bool oky1 = (iy1 >= 0) & (iy1 < NH);

        const int xc0 = min(max(ix0, 0), NW - 1);
        const int xc1 = min(max(ix1, 0), NW - 1);
        const int yc0 = min(max(iy0, 0), NH - 1);
        const int yc1 = min(max(iy1, 0), NH - 1);

        const int i00 = (yc0 << 8) | xc0;
        const int i01 = (yc0 << 8) | xc1;
        const int i10 = (yc1 << 8) | xc0;
        const int i11 = (yc1 << 8) | xc1;

        // Fold validity into the bilinear weights: no branches, clamped
        // gather reads a safe address, weight 0 kills the contribution.
        const float wx0 = 1.0f - wx, wy0 = 1.0f - wy;
        const float w00 = (okx0 && oky0) ? wx0 * wy0 : 0.0f;
        const float w01 = (okx1 && oky0) ? wx  * wy0 : 0.0f;
        const float w10 = (okx0 && oky1) ? wx0 * wy  : 0.0f;
        const float w11 = (okx1 && oky1) ? wx  * wy  : 0.0f;

        const int oidx = (y << 8) | x;

        // One coordinate computation amortized over all 4 channels:
        // 16 gathers + 4 NT stores per pixel-group.
#pragma unroll
        for (int c = 0; c < NC; ++c) {
            const float* __restrict__ cb = sbase + c * NHW;
            float v =        w00 * cb[i00];
            v = fmaf(w01, cb[i01], v);
            v = fmaf(w10, cb[i10], v);
            v = fmaf(w11, cb[i11], v);
            __builtin_nontemporal_store(v, obase + c * NHW + oidx);
        }
    }
}

extern "C" void kernel_launch(void* const* d_in, const int* in_sizes, int n_in,
                              void* d_out, int out_size, void* d_ws, size_t ws_size,
                              hipStream_t stream) {
    (void)in_sizes; (void)n_in; (void)out_size; (void)d_ws; (void)ws_size;
    const float* hmat = (const float*)d_in[0];   // (4,32,3,3) fp32
    const float* src  = (const float*)d_in[1];   // (4,32,4,256,256) fp32
    float*       out  = (float*)d_out;           // (4,32,4,256,256) fp32

    dim3 grid(NH / ROWS, NBD);   // (64, 128)
    dim3 block(NW);              // 256 threads = 8 wave32
    ApplyHomography_warp_kernel<<<grid, block, 0, stream>>>(hmat, src, out);
}